// Dice_58909771432702
// MI455X (gfx1250) — compile-verified
//
#include <hip/hip_runtime.h>

typedef int   iv4 __attribute__((ext_vector_type(4)));
typedef float v2f __attribute__((ext_vector_type(2)));
typedef float v8f __attribute__((ext_vector_type(8)));

#define NUM_CLASSES 5
#define NBATCH 4
#define VOX_PER_BATCH (64 * 256 * 256)          // 4,194,304 voxels per batch
#define VEC_PER_BATCH (VOX_PER_BATCH / 4)       // 1,048,576 int4 per batch
#define THREADS 256
#define BLOCKS_PER_BATCH 256
#define ITERS (VEC_PER_BATCH / (THREADS * BLOCKS_PER_BATCH))  // 16 int4 = 64 voxels/thread
#define NCOUNT (NBATCH * 3 * NUM_CLASSES)       // 60 global accumulators

// ---------------------------------------------------------------------------
// Zero the 60 global accumulators (d_ws is poisoned, and accumulation must
// start from zero on every launch for determinism).
// ---------------------------------------------------------------------------
__global__ void dice_init_kernel(unsigned* __restrict__ g) {
    if (threadIdx.x < NCOUNT) g[threadIdx.x] = 0u;
}

// ---------------------------------------------------------------------------
// Streaming count kernel. Memory-bound: 134 MB @ 23.3 TB/s ~ 5.8 us floor.
// Per voxel pair: ~12 VALU ops using packed 12-bit-per-class u64 counters:
//   accP += 1<<(12p) ; accT += 1<<(12t) ; accI += (1<<(12p)) & (1<<(12t))
// (the AND is nonzero exactly when p==t, landing in class p's field).
// Each thread touches 64 voxels -> fields <= 64, single extraction at end.
// ---------------------------------------------------------------------------
__global__ __launch_bounds__(THREADS)
void dice_count_kernel(const int* __restrict__ pred,
                       const int* __restrict__ targ,
                       unsigned* __restrict__ g) {
    const int b = blockIdx.y;
    const iv4* __restrict__ p4 = (const iv4*)(pred + (size_t)b * VOX_PER_BATCH);
    const iv4* __restrict__ t4 = (const iv4*)(targ + (size_t)b * VOX_PER_BATCH);

    unsigned idx = blockIdx.x * THREADS + threadIdx.x;
    unsigned long long accP = 0ull, accT = 0ull, accI = 0ull;

    #pragma unroll 4
    for (int it = 0; it < ITERS; ++it) {
        // Non-temporal: data is touched exactly once, keep it out of caches.
        iv4 p = __builtin_nontemporal_load(&p4[idx]);
        iv4 q = __builtin_nontemporal_load(&t4[idx]);
        #pragma unroll
        for (int e = 0; e < 4; ++e) {
            unsigned long long a = 1ull << (unsigned)(p[e] * 12);
            unsigned long long c = 1ull << (unsigned)(q[e] * 12);
            accP += a;
            accT += c;
            accI += (a & c);
        }
        idx += THREADS * BLOCKS_PER_BATCH;
    }

    // Unpack the 15 per-thread counters (each <= 64, fits 12 bits easily).
    unsigned cnt[3 * NUM_CLASSES];
    #pragma unroll
    for (int c = 0; c < NUM_CLASSES; ++c) {
        cnt[c]                   = (unsigned)(accP >> (12 * c)) & 0xFFFu;
        cnt[NUM_CLASSES + c]     = (unsigned)(accT >> (12 * c)) & 0xFFFu;
        cnt[2 * NUM_CLASSES + c] = (unsigned)(accI >> (12 * c)) & 0xFFFu;
    }

    // wave32 butterfly reduction (5 steps), then per-wave leader -> LDS.
    #pragma unroll
    for (int i = 0; i < 3 * NUM_CLASSES; ++i) {
        #pragma unroll
        for (int off = 16; off > 0; off >>= 1)
            cnt[i] += (unsigned)__shfl_xor((int)cnt[i], off, 32);
    }

    __shared__ unsigned s_cnt[3 * NUM_CLASSES];
    if (threadIdx.x < 3 * NUM_CLASSES) s_cnt[threadIdx.x] = 0u;
    __syncthreads();
    if ((threadIdx.x & 31) == 0) {
        #pragma unroll
        for (int i = 0; i < 3 * NUM_CLASSES; ++i)
            atomicAdd(&s_cnt[i], cnt[i]);
    }
    __syncthreads();
    if (threadIdx.x < 3 * NUM_CLASSES)
        atomicAdd(&g[b * (3 * NUM_CLASSES) + threadIdx.x], s_cnt[threadIdx.x]);
}

// ---------------------------------------------------------------------------
// Final kernel: dice[c] = 0.25 * sum_b r[b,c],  r = (den==0) ? 1 : 2I/den.
// The batch-mean is a 16x16x4 f32 WMMA: A[m,k] = r(batch k, class m) in the
// documented 32-bit A layout (lane L<16 -> A[L,{0,1}], lane L+16 -> A[L,{2,3}]),
// B = 0.25 broadcast (layout-independent constant). C row m = dice[m]; thread
// 0 (N=0 column) holds dice[0..7] in its 8 accumulator VGPRs.
// One wave, straight-line code before the WMMA -> EXEC is all ones.
// ---------------------------------------------------------------------------
__global__ void dice_final_kernel(const unsigned* __restrict__ g,
                                  float* __restrict__ out) {
    const int lane = threadIdx.x;           // 0..31
    const int row  = lane & 15;             // class index m for this lane
    const int bb   = (lane < 16) ? 0 : 2;   // batches {0,1} or {2,3}
    const int c    = (row < NUM_CLASSES) ? row : 0;  // clamp (branchless)
    const float keep = (row < NUM_CLASSES) ? 1.0f : 0.0f;

    v2f a;
    #pragma unroll
    for (int k = 0; k < 2; ++k) {
        const int bsel = bb + k;
        const float P = (float)g[bsel * (3 * NUM_CLASSES) + c];
        const float T = (float)g[bsel * (3 * NUM_CLASSES) + NUM_CLASSES + c];
        const float I = (float)g[bsel * (3 * NUM_CLASSES) + 2 * NUM_CLASSES + c];
        const float den = P + T;
        const float r = (den == 0.0f) ? 1.0f : (2.0f * I) / den;
        a[k] = r * keep;
    }

#if __has_builtin(__builtin_amdgcn_wmma_f32_16x16x4_f32)
    v2f bmat = {0.25f, 0.25f};   // B[k,n] = 0.25 everywhere
    v8f c8 = {};
    c8 = __builtin_amdgcn_wmma_f32_16x16x4_f32(
        /*neg_a=*/false, a, /*neg_b=*/false, bmat,
        /*c_mod=*/(short)0, c8, /*reuse_a=*/false, /*reuse_b=*/false);
    if (lane == 0) {
        out[0] = c8[0]; out[1] = c8[1]; out[2] = c8[2];
        out[3] = c8[3]; out[4] = c8[4];
    }
#else
    // Fallback: scalar mean (no WMMA available on this toolchain).
    __shared__ float s_r[2][16];
    s_r[lane >> 4][row] = a[0] + a[1];   // r(b0)+r(b1) or r(b2)+r(b3)
    __syncthreads();
    if (lane < NUM_CLASSES)
        out[lane] = 0.25f * (s_r[0][lane] + s_r[1][lane]);
#endif
}

// ---------------------------------------------------------------------------
extern "C" void kernel_launch(void* const* d_in, const int* in_sizes, int n_in,
                              void* d_out, int out_size, void* d_ws, size_t ws_size,
                              hipStream_t stream) {
    const int* pred = (const int*)d_in[0];
    const int* targ = (const int*)d_in[1];
    float* out      = (float*)d_out;
    unsigned* g     = (unsigned*)d_ws;    // 60 x u32 accumulators

    dice_init_kernel<<<1, 64, 0, stream>>>(g);

    dim3 grid(BLOCKS_PER_BATCH, NBATCH, 1);
    dice_count_kernel<<<grid, THREADS, 0, stream>>>(pred, targ, g);

    dice_final_kernel<<<1, 32, 0, stream>>>(g, out);
}